// VTMDownsamplerCorrect_85547158602078
// MI455X (gfx1250) — compile-verified
//
#include <hip/hip_runtime.h>

// ---------------------------------------------------------------------------
// VTM downsampler, fused single pass, exact integer arithmetic in fp32 via
// V_WMMA_F32_16X16X4_F32.  OUT_tile(16x16) = C * IN_tile(16x16) * C^T.
// Memory-bound workload (~50 MB stores, ~3.5 MB unique cached input);
// one wave32 per 16x16 output tile, 8 WMMAs per tile.
// ---------------------------------------------------------------------------

typedef float v2f __attribute__((ext_vector_type(2)));
typedef float v8f __attribute__((ext_vector_type(8)));

namespace {

constexpr int H  = 1080, W  = 1920;
constexpr int HO = 540,  WO = 960;
constexpr int TILES_X = WO / 16;          // 60
constexpr int TILES_Y = (HO + 15) / 16;   // 34 (last tile partial in rows)
constexpr int WAVES   = 8;                // waves (tiles) per block

// Banded 16x16 coefficient matrix: C[j][t] = F[phase(j)][t - (j>>2)],
// phase(j) in {0,4,8,12} -> only 4 distinct VTM filter rows are ever used.
struct CoefTable {
  float c[16][16];
  constexpr CoefTable() : c{} {
    const float F[4][12] = {
        {0, 0, 0, 0,   0, 128,   0,   0, 0, 0, 0, 0},   // phase 0
        {0, 0, 0, 5, -18, 114,  36, -10, 1, 0, 0, 0},   // phase 4
        {0, 0, 0, 4, -19,  79,  79, -19, 4, 0, 0, 0},   // phase 8
        {0, 0, 0, 1, -10,  36, 114, -18, 5, 0, 0, 0}};  // phase 12
    for (int j = 0; j < 16; ++j)
      for (int t = 0; t < 16; ++t) {
        const int k = t - (j >> 2);
        c[j][t] = (k >= 0 && k < 12) ? F[j & 3][k] : 0.0f;
      }
  }
};
__device__ constexpr CoefTable COEF{};

__global__ __launch_bounds__(WAVES * 32)
void vtm_downsample_wmma(const float* __restrict__ x, float* __restrict__ out,
                         int total_tiles) {
  // Per-wave 16x16 staging for T = IN * C^T, padded row stride 20 floats so
  // the (row g | row g+8) half-wave accesses hit disjoint bank ranges.
  __shared__ float Tlds[WAVES][16][20];

  const int lane = threadIdx.x & 31;
  const int wid  = threadIdx.x >> 5;
  const int tile = blockIdx.x * WAVES + wid;
  if (tile >= total_tiles) return;  // uniform per wave -> EXEC stays all-ones

  const int tiles_per_img = TILES_X * TILES_Y;
  const int bc = tile / tiles_per_img;
  const int t2 = tile - bc * tiles_per_img;
  const int ty = t2 / TILES_X;
  const int tx = t2 - ty * TILES_X;

  const float* img  = x   + (size_t)bc * ((size_t)H * W);
  float*       oimg = out + (size_t)bc * ((size_t)HO * WO);

  const int m  = lane & 15;   // A-matrix row / B-matrix column
  const int hi = lane >> 4;   // half-wave select (K offset +2 in A/B layout)

  // Input row this lane supplies for GEMM1's A operand (edge-clamped).
  int ri = 4 * ty - 5 + m;
  ri = ri < 0 ? 0 : (ri > H - 1 ? H - 1 : ri);
  const float* rowp  = img + (size_t)ri * W;
  const int    cbase = 4 * tx - 5;

  // Coefficient operand: identical register image serves as B of GEMM1
  // (C^T chunk) and A of GEMM2 (C chunk), per the mirrored wave32 layouts:
  //   A: vgpr v, lane -> A[m][4kk + 2*hi + v]
  //   B: vgpr v, lane -> B[2*hi + v][m] = C[m][4kk + 2*hi + v]
  v2f coef[4];
#pragma unroll
  for (int kk = 0; kk < 4; ++kk) {
    const int w0 = 4 * kk + 2 * hi;
    coef[kk].x = COEF.c[m][w0];
    coef[kk].y = COEF.c[m][w0 + 1];
  }

  // ---- GEMM1 (horizontal pass): T = IN(16x16) x C^T, 4 x K=4 WMMA --------
  v8f t = {0.f, 0.f, 0.f, 0.f, 0.f, 0.f, 0.f, 0.f};
#pragma unroll
  for (int kk = 0; kk < 4; ++kk) {
    int c0 = cbase + 4 * kk + 2 * hi;
    int c1 = c0 + 1;
    c0 = c0 < 0 ? 0 : (c0 > W - 1 ? W - 1 : c0);
    c1 = c1 < 0 ? 0 : (c1 > W - 1 ? W - 1 : c1);
    v2f a;
    a.x = rowp[c0];
    a.y = rowp[c1];
    t = __builtin_amdgcn_wmma_f32_16x16x4_f32(false, a, false, coef[kk],
                                              (short)0, t, false, false);
  }
  // All T entries are exact integers (|T| <= ~52020 < 2^24) -> reference's
  // trunc() is a mathematical no-op.

  // ---- Re-layout T: C/D format -> B format via per-wave LDS ---------------
#pragma unroll
  for (int g = 0; g < 8; ++g)
    Tlds[wid][g + 8 * hi][m] = t[g];   // D layout: vgpr g = rows g / g+8

  v2f tb[4];
#pragma unroll
  for (int kk = 0; kk < 4; ++kk) {     // B layout: vgpr v = rows 4kk+2hi+v
    tb[kk].x = Tlds[wid][4 * kk + 2 * hi][m];
    tb[kk].y = Tlds[wid][4 * kk + 2 * hi + 1][m];
  }

  // ---- GEMM2 (vertical pass): O = C x T, 4 x K=4 WMMA ---------------------
  v8f o = {0.f, 0.f, 0.f, 0.f, 0.f, 0.f, 0.f, 0.f};
#pragma unroll
  for (int kk = 0; kk < 4; ++kk)
    o = __builtin_amdgcn_wmma_f32_16x16x4_f32(false, coef[kk], false, tb[kk],
                                              (short)0, o, false, false);

  // ---- Epilogue: exact (acc + 8192) >> 14, clip to [0,255], store ---------
  float res[8];
#pragma unroll
  for (int g = 0; g < 8; ++g) {
    const float v = floorf((o[g] + 8192.0f) * (1.0f / 16384.0f));
    res[g] = fminf(fmaxf(v, 0.0f), 255.0f);   // v_max_num / v_min_num
  }

  const int r0  = 16 * ty + 8 * hi;
  float* obase  = oimg + (size_t)r0 * WO + (16 * tx + m);  // cols always valid
  if (ty != TILES_Y - 1) {
    // Common path (33/34 tiles): fully in-bounds, unconditional NT stores,
    // no EXEC-mask churn.
#pragma unroll
    for (int g = 0; g < 8; ++g)
      __builtin_nontemporal_store(res[g], obase + (size_t)g * WO);
  } else {
    // Last row-tile: rows 528..543, only r < 540 valid.
#pragma unroll
    for (int g = 0; g < 8; ++g)
      if (r0 + g < HO)
        __builtin_nontemporal_store(res[g], obase + (size_t)g * WO);
  }
}

}  // namespace

extern "C" void kernel_launch(void* const* d_in, const int* in_sizes, int n_in,
                              void* d_out, int out_size, void* d_ws,
                              size_t ws_size, hipStream_t stream) {
  (void)n_in; (void)out_size; (void)d_ws; (void)ws_size;
  const float* x   = (const float*)d_in[0];
  float*       out = (float*)d_out;

  const int nbc         = in_sizes[0] / (H * W);        // 8*3 = 24 planes
  const int total_tiles = nbc * TILES_X * TILES_Y;      // 48960
  const int blocks      = (total_tiles + WAVES - 1) / WAVES;  // 6120

  vtm_downsample_wmma<<<blocks, WAVES * 32, 0, stream>>>(x, out, total_tiles);
}